// PGCN_40707700031795
// MI455X (gfx1250) — compile-verified
//
#include <hip/hip_runtime.h>

#define NG   12331
#define ND   5000
#define DIN  512
#define H1   64
#define H2   32
#define E_GG 1000000
#define E_GD 400000
#define E_DD 100000

typedef __attribute__((ext_vector_type(2))) float v2f;
typedef __attribute__((ext_vector_type(8))) float v8f;

__global__ void k_zero(float* __restrict__ p, int n) {
    int i = blockIdx.x * blockDim.x + threadIdx.x;
    if (i < n) p[i] = 0.0f;
}

__global__ void k_zero4(float4* __restrict__ p, int n4) {
    int i = blockIdx.x * blockDim.x + threadIdx.x;
    if (i < n4) p[i] = make_float4(0.f, 0.f, 0.f, 0.f);
}

__global__ void k_count(const int* __restrict__ idx, int n, float* __restrict__ deg) {
    int i = blockIdx.x * blockDim.x + threadIdx.x;
    if (i < n) atomicAdd(&deg[idx[i]], 1.0f);
}

__global__ void k_invsqrt(float* __restrict__ p, int n) {
    int i = blockIdx.x * blockDim.x + threadIdx.x;
    if (i < n) p[i] = rsqrtf(fmaxf(p[i], 1.0f));
}

// agg[dst[e]][d..d+3] += h[src[e]][d..d+3] * inv_src[src[e]]   (F = 1<<logF dims)
// one thread per (edge, 4 dims): 1x b128 load + 4x global_atomic_add_f32
__global__ void k_scatter4(const float* __restrict__ h, const int* __restrict__ src,
                           const int* __restrict__ dst, const float* __restrict__ inv_src,
                           float* __restrict__ agg, int E, int logF) {
    int t = blockIdx.x * blockDim.x + threadIdx.x;
    int qF = logF - 2;                       // quads per node row
    int e = t >> qF;
    int d = (t & ((1 << qF) - 1)) << 2;
    if (e < E) {
        int s = src[e];
        float is = inv_src[s];
        float4 v = *reinterpret_cast<const float4*>(h + ((long long)s << logF) + d);
        float* ap = agg + ((long long)dst[e] << logF) + d;
        atomicAdd(ap + 0, v.x * is);
        atomicAdd(ap + 1, v.y * is);
        atomicAdd(ap + 2, v.z * is);
        atomicAdd(ap + 3, v.w * is);
    }
}

// C[M,N] (+)= relu?( (rowscale ⊙ A[M,K]) @ W[N,K]^T + bias )  via V_WMMA_F32_16X16X4_F32.
// One wave computes one 16x16 C tile; 8 waves per block.
// Out-of-range rows/cols use clamped (safe, finite) addresses with NO in-loop masking:
// WMMA element (m,n) depends only on A row m / B col n, and guarded stores never emit
// the out-of-range elements, so clamped-row garbage cannot contaminate valid outputs.
template <bool HAS_RS, bool ACC, bool RELU>
__global__ void k_gemm_wmma(const float* __restrict__ A, const float* __restrict__ W,
                            const float* __restrict__ bias, const float* __restrict__ rowscale,
                            float* __restrict__ C, int M, int N, int K) {
    int wave = threadIdx.x >> 5;
    int lane = threadIdx.x & 31;
    int nt = (N + 15) >> 4;
    int mt = (M + 15) >> 4;
    int tile = blockIdx.x * (blockDim.x >> 5) + wave;
    if (tile >= mt * nt) return;             // wave-uniform exit (EXEC stays all-1s)
    int tm = tile / nt, tn = tile % nt;
    int row0 = tm << 4, col0 = tn << 4;

    int half = lane >> 4;                    // 0: this lane covers K{0,1}; 1: K{2,3}
    int lm   = lane & 15;

    int ma = row0 + lm;                      // A row handled by this lane
    int nb = col0 + lm;                      // W row (= C column) handled by this lane
    bool mok = ma < M;
    bool nok = nb < N;
    int mac = mok ? ma : (M - 1);            // clamped, load-safe
    int nbc = nok ? nb : (N - 1);
    float rs = HAS_RS ? rowscale[mac] : 1.0f;

    // accumulator init: bias (+ existing C when accumulating)
    v8f c;
    int mb = row0 + (half << 3);             // lanes 16-31 hold tile rows 8..15
    float bn = (bias != nullptr) ? bias[nbc] : 0.0f;
#pragma unroll
    for (int r = 0; r < 8; ++r) {
        int m = mb + r;
        float v = bn;
        if (ACC) { if (nok && m < M) v += C[(long long)m * N + nbc]; }
        c[r] = v;
    }

    const float* Arow = A + (long long)mac * K + (half << 1);
    const float* Wrow = W + (long long)nbc * K + (half << 1);
    for (int k0 = 0; k0 < K; k0 += 4) {
        v2f a = *reinterpret_cast<const v2f*>(Arow + k0);   // ISA 16x4 f32 A: v0=K{0,2}, v1=K{1,3}
        v2f b = *reinterpret_cast<const v2f*>(Wrow + k0);   // B 4x16 mirrored layout
        if (HAS_RS) { a.x *= rs; a.y *= rs; }
        c = __builtin_amdgcn_wmma_f32_16x16x4_f32(
            /*neg_a=*/false, a, /*neg_b=*/false, b,
            /*c_mod=*/(short)0, c, /*reuse_a=*/false, /*reuse_b=*/false);
    }

#pragma unroll
    for (int r = 0; r < 8; ++r) {
        int m = mb + r;
        if (m < M && nok) {
            float v = c[r];
            if (RELU) v = fmaxf(v, 0.0f);
            C[(long long)m * N + nbc] = v;
        }
    }
}

static inline int cdiv_i(long long a, long long b) { return (int)((a + b - 1) / b); }

extern "C" void kernel_launch(void* const* d_in, const int* in_sizes, int n_in,
                              void* d_out, int out_size, void* d_ws, size_t ws_size,
                              hipStream_t stream) {
    const float* feat_g = (const float*)d_in[0];
    const float* feat_d = (const float*)d_in[1];
    const int* gg_src = (const int*)d_in[2];
    const int* gg_dst = (const int*)d_in[3];
    const int* gd_src = (const int*)d_in[4];
    const int* gd_dst = (const int*)d_in[5];
    const int* dd_src = (const int*)d_in[6];
    const int* dd_dst = (const int*)d_in[7];
    const float* Wg   = (const float*)d_in[8];
    const float* Wd   = (const float*)d_in[9];
    const float* W1   = (const float*)d_in[10];
    const float* b1   = (const float*)d_in[11];
    const float* W2   = (const float*)d_in[12];
    const float* b2   = (const float*)d_in[13];
    const float* Wout = (const float*)d_in[14];
    float* out = (float*)d_out;

    // workspace layout (floats), every block padded to 32 floats (128 B) for b64/b128 alignment
    float* ws = (float*)d_ws;
    size_t off = 0;
    auto alloc = [&](size_t n) { float* p = ws + off; off += (n + 31) & ~(size_t)31; return p; };
    float* inv_ggs = alloc(NG);   // deg(gg_src) over NG  -> ^-1/2
    float* inv_ggd = alloc(NG);   // deg(gg_dst) over NG
    float* inv_gds = alloc(NG);   // deg(gd_src) over NG
    float* inv_gdd = alloc(ND);   // deg(gd_dst) over ND
    float* inv_dds = alloc(ND);   // deg(dd_src) over ND
    float* inv_ddd = alloc(ND);   // deg(dd_dst) over ND
    float* hgE  = alloc((size_t)NG * H1);
    float* hdE  = alloc((size_t)ND * H1);
    float* aggA = alloc((size_t)NG * H1);   // gene <- gene_gene
    float* aggB = alloc((size_t)NG * H1);   // gene <- disease_gene
    float* aggC = alloc((size_t)ND * H1);   // dis  <- gene_disease
    float* aggD = alloc((size_t)ND * H1);   // dis  <- disease_disease
    float* hg1  = alloc((size_t)NG * H1);
    float* hd1  = alloc((size_t)ND * H1);
    float* hg2  = alloc((size_t)NG * H2);
    float* hd2  = alloc((size_t)ND * H2);
    float* Tbuf = alloc((size_t)NG * H2);

    const int TB = 256;
    auto gemm = [&](const float* A, const float* W, const float* bias, const float* rsc,
                    float* C, int M, int N, int K, int acc_relu) {
        long long tiles = (long long)cdiv_i(M, 16) * cdiv_i(N, 16);
        int blocks = cdiv_i(tiles, TB / 32);
        if (rsc == nullptr)
            k_gemm_wmma<false, false, false><<<blocks, TB, 0, stream>>>(A, W, bias, rsc, C, M, N, K);
        else if (!acc_relu)
            k_gemm_wmma<true, false, false><<<blocks, TB, 0, stream>>>(A, W, bias, rsc, C, M, N, K);
        else
            k_gemm_wmma<true, true, true><<<blocks, TB, 0, stream>>>(A, W, bias, rsc, C, M, N, K);
    };

    // --- degrees -> D^{-1/2} ---
    k_zero<<<cdiv_i(3 * NG, TB), TB, 0, stream>>>(inv_ggs, 3 * NG);   // 3 NG arrays, padded to 12352 each; last 42 of the 3rd not covered
    k_zero<<<cdiv_i(3 * ND + 96, TB), TB, 0, stream>>>(inv_gdd, 3 * ND + 96); // 3 padded ND arrays
    k_count<<<cdiv_i(E_GG, TB), TB, 0, stream>>>(gg_src, E_GG, inv_ggs);
    k_count<<<cdiv_i(E_GG, TB), TB, 0, stream>>>(gg_dst, E_GG, inv_ggd);
    k_count<<<cdiv_i(E_GD, TB), TB, 0, stream>>>(gd_src, E_GD, inv_gds);
    k_count<<<cdiv_i(E_GD, TB), TB, 0, stream>>>(gd_dst, E_GD, inv_gdd);
    k_count<<<cdiv_i(E_DD, TB), TB, 0, stream>>>(dd_src, E_DD, inv_dds);
    k_count<<<cdiv_i(E_DD, TB), TB, 0, stream>>>(dd_dst, E_DD, inv_ddd);
    k_invsqrt<<<cdiv_i(3 * NG, TB), TB, 0, stream>>>(inv_ggs, 3 * NG);
    k_invsqrt<<<cdiv_i(3 * ND + 96, TB), TB, 0, stream>>>(inv_gdd, 3 * ND + 96);

    // --- embeddings: h = feat @ Wemb.T ---
    gemm(feat_g, Wg, nullptr, nullptr, hgE, NG, H1, DIN, 0);
    gemm(feat_d, Wd, nullptr, nullptr, hdE, ND, H1, DIN, 0);

    // --- hetero layer (aggregation always on 64-dim inputs; Fout = 64 then 32) ---
    auto layer = [&](const float* hg_in, const float* hd_in,
                     const float* Wl, const float* bl, int Fout,
                     float* outg, float* outd) {
        // zero the four aggregation buffers (laid out back to back with 128B pads)
        auto z4 = [&](float* p, size_t n) {
            k_zero4<<<cdiv_i(n / 4, TB), TB, 0, stream>>>((float4*)p, (int)(n / 4));
        };
        z4(aggA, (size_t)NG * H1); z4(aggB, (size_t)NG * H1);
        z4(aggC, (size_t)ND * H1); z4(aggD, (size_t)ND * H1);
        long long tq = (long long)H1 / 4;
        k_scatter4<<<cdiv_i((long long)E_GG * tq, TB), TB, 0, stream>>>(hg_in, gg_src, gg_dst, inv_ggs, aggA, E_GG, 6);
        k_scatter4<<<cdiv_i((long long)E_GD * tq, TB), TB, 0, stream>>>(hd_in, gd_dst, gd_src, inv_gdd, aggB, E_GD, 6);
        k_scatter4<<<cdiv_i((long long)E_GD * tq, TB), TB, 0, stream>>>(hg_in, gd_src, gd_dst, inv_gds, aggC, E_GD, 6);
        k_scatter4<<<cdiv_i((long long)E_DD * tq, TB), TB, 0, stream>>>(hd_in, dd_src, dd_dst, inv_dds, aggD, E_DD, 6);
        // g_out = relu( (aggA*ggd^-1/2)@W[0].T + b0 + (aggB*gds^-1/2)@W[2].T + b2 )
        gemm(aggA, Wl + 0 * (size_t)Fout * H1, bl + 0 * Fout, inv_ggd, outg, NG, Fout, H1, 0);
        gemm(aggB, Wl + 2 * (size_t)Fout * H1, bl + 2 * Fout, inv_gds, outg, NG, Fout, H1, 1);
        // d_out = relu( (aggC*gdd^-1/2)@W[1].T + b1 + (aggD*ddd^-1/2)@W[3].T + b3 )
        gemm(aggC, Wl + 1 * (size_t)Fout * H1, bl + 1 * Fout, inv_gdd, outd, ND, Fout, H1, 0);
        gemm(aggD, Wl + 3 * (size_t)Fout * H1, bl + 3 * Fout, inv_ddd, outd, ND, Fout, H1, 1);
    };

    layer(hgE, hdE, W1, b1, H1, hg1, hd1);
    layer(hg1, hd1, W2, b2, H2, hg2, hd2);

    // --- output: (hg2 @ Wout.T) @ hd2.T ---
    gemm(hg2, Wout, nullptr, nullptr, Tbuf, NG, H2, H2, 0);
    gemm(Tbuf, hd2, nullptr, nullptr, out, NG, ND, H2, 0);

    (void)in_sizes; (void)n_in; (void)out_size; (void)ws_size;
}